// GatedLayer_7859790152274
// MI455X (gfx1250) — compile-verified
//
#include <hip/hip_runtime.h>
#include <math.h>

#define DEG      32
#define NFEAT    128
#define NCLASSES 40
#define NB_RED   128     // partial-reduction blocks; 128 = 2 WMMA chunks of 64
#define LN_EPS   1e-5f
#define CLAMP_MIN 1e-5f

typedef __attribute__((ext_vector_type(2))) float v2f;
typedef __attribute__((ext_vector_type(8))) float v8f;

// ---------------------------------------------------------------- K1: argmax
__global__ void pred_kernel(const float* __restrict__ logits,
                            int* __restrict__ pred, int n_nodes) {
    int n = blockIdx.x * blockDim.x + threadIdx.x;
    if (n >= n_nodes) return;
    const float* row = logits + (size_t)n * NCLASSES;
    float best = row[0];
    int bi = 0;
    #pragma unroll
    for (int c = 1; c < NCLASSES; ++c) {
        float v = row[c];
        if (v > best) { best = v; bi = c; }   // first-max tie break (jnp.argmax)
    }
    pred[n] = bi;
}

// ------------------------------------------- K2: f1 (agreement) + f2 (entropy)
// one wave32 per node; lane = one of the 32 in-edges
__global__ void f1f2_kernel(const int* __restrict__ pred,
                            const int* __restrict__ src,
                            float* __restrict__ f1, float* __restrict__ f2,
                            int n_nodes) {
    int lane = threadIdx.x & 31;
    int wid  = threadIdx.x >> 5;
    int n = blockIdx.x * (blockDim.x >> 5) + wid;
    if (n >= n_nodes) return;                 // uniform per wave

    int my_nbr = pred[src[n * DEG + lane]];
    int p_self = pred[n];

    unsigned long long m = __ballot(my_nbr == p_self);
    float f1v = (float)__popcll(m) * (1.0f / DEG);

    // lane L counts neighbors predicting class L (and class L+32 if < 40)
    int c0 = 0, c1 = 0;
    int cls1 = lane + 32;
    #pragma unroll
    for (int j = 0; j < DEG; ++j) {
        int v = __shfl(my_nbr, j);            // wave32 broadcast
        c0 += (v == lane);
        c1 += (v == cls1);
    }
    const float inv = 1.0f / DEG;
    float p0 = fmaxf((float)c0 * inv, CLAMP_MIN);
    float t  = -p0 * logf(p0);
    if (cls1 < NCLASSES) {
        float p1 = fmaxf((float)c1 * inv, CLAMP_MIN);
        t += -p1 * logf(p1);
    }
    #pragma unroll
    for (int o = 16; o > 0; o >>= 1) t += __shfl_xor(t, o);

    if (lane == 0) { f1[n] = f1v; f2[n] = t; }
}

// ----------------------------- K3: per-block partial sums for global LayerNorm
__global__ void partial_kernel(const float* __restrict__ f1,
                               const float* __restrict__ f2,
                               float* __restrict__ partials, int n_nodes) {
    __shared__ float sm[4 * 256];
    int t = threadIdx.x;
    float s1 = 0.f, q1 = 0.f, s2 = 0.f, q2 = 0.f;
    for (int i = blockIdx.x * blockDim.x + t; i < n_nodes;
         i += gridDim.x * blockDim.x) {
        float a = f1[i], b = f2[i];
        s1 += a; q1 += a * a; s2 += b; q2 += b * b;
    }
    sm[t] = s1; sm[256 + t] = q1; sm[512 + t] = s2; sm[768 + t] = q2;
    __syncthreads();
    for (int o = 128; o > 0; o >>= 1) {
        if (t < o) {
            sm[t]       += sm[t + o];
            sm[256 + t] += sm[256 + t + o];
            sm[512 + t] += sm[512 + t + o];
            sm[768 + t] += sm[768 + t + o];
        }
        __syncthreads();
    }
    if (t == 0) {
        partials[0 * NB_RED + blockIdx.x] = sm[0];
        partials[1 * NB_RED + blockIdx.x] = sm[256];
        partials[2 * NB_RED + blockIdx.x] = sm[512];
        partials[3 * NB_RED + blockIdx.x] = sm[768];
    }
}

// ------------- K4: single-wave final reduction via V_WMMA_F32_16X16X4_F32
// D = ones(16x4) x B(4x16) + C  => every D row = column sums of B.
// Two chained WMMAs consume 128 partials per stream; butterfly*0.5 finishes.
__global__ void stats_kernel(const float* __restrict__ partials,
                             float* __restrict__ stats, int n_nodes) {
    int lane = threadIdx.x & 31;
    v2f ones; ones.x = 1.0f; ones.y = 1.0f;
    float tot[4];
    #pragma unroll
    for (int s = 0; s < 4; ++s) {
        v8f acc = {};
        #pragma unroll
        for (int c = 0; c < NB_RED / 64; ++c) {
            const float* base = partials + s * NB_RED + c * 64;
            v2f b;
            b.x = base[lane];
            b.y = base[32 + lane];
            acc = __builtin_amdgcn_wmma_f32_16x16x4_f32(
                false, ones, false, b, (short)0, acc, false, false);
        }
        float v = acc[0];                      // row M=0 (lanes 0-15), M=8 dup
        #pragma unroll
        for (int o = 16; o > 0; o >>= 1) v += __shfl_xor(v, o);
        tot[s] = v * 0.5f;                     // halves were duplicates
    }
    if (lane == 0) {
        float invn = 1.0f / (float)n_nodes;
        float m1 = tot[0] * invn;
        float v1 = tot[1] * invn - m1 * m1;    // ddof=0, matches jnp.var
        float m2 = tot[2] * invn;
        float v2 = tot[3] * invn - m2 * m2;
        stats[0] = m1;
        stats[1] = rsqrtf(v1 + LN_EPS);
        stats[2] = m2;
        stats[3] = rsqrtf(v2 + LN_EPS);
    }
}

// --------------------------------------------------------- K5: z and gate
__global__ void zgate_kernel(const float* __restrict__ f1,
                             const float* __restrict__ f2,
                             const float* __restrict__ old_z,
                             const float* __restrict__ tau1,
                             const float* __restrict__ tau2,
                             const float* __restrict__ stats,
                             float* __restrict__ z_out,
                             float* __restrict__ gate, int n_nodes) {
    int n = blockIdx.x * blockDim.x + threadIdx.x;
    if (n >= n_nodes) return;
    float m1 = stats[0], r1 = stats[1], m2 = stats[2], r2 = stats[3];
    float a = (f1[n] - m1) * r1 - tau1[0];   // sigmoid(-(nf - tau)) = 1/(1+e^a)
    float b = (f2[n] - m2) * r2 - tau2[0];
    float z = (1.0f / (1.0f + expf(a))) * (1.0f / (1.0f + expf(b)));
    z_out[n] = z;
    gate[n]  = fminf(old_z[n], z);
}

// ----------------- K6: gather-sum aggregation + epilogue (wave32 per node)
// lane L owns features [4L,4L+4): no cross-lane reduction needed.
__global__ void agg_kernel(const float* __restrict__ h,
                           const int* __restrict__ src,
                           const float* __restrict__ norm,
                           const float* __restrict__ gate,
                           float* __restrict__ new_h, int n_nodes) {
    int lane = threadIdx.x & 31;
    int wid  = threadIdx.x >> 5;
    int n = blockIdx.x * (blockDim.x >> 5) + wid;
    if (n >= n_nodes) return;                 // uniform per wave

    int idx = src[n * DEG + lane];            // lane L holds neighbor L's index
    float4 acc = make_float4(0.f, 0.f, 0.f, 0.f);

    #pragma unroll
    for (int k = 0; k < DEG; k += 4) {        // keep 4 b128 gathers in flight
        int s0 = __shfl(idx, k + 0);
        int s1 = __shfl(idx, k + 1);
        int s2 = __shfl(idx, k + 2);
        int s3 = __shfl(idx, k + 3);
        float4 r0 = *(const float4*)(h + (size_t)s0 * NFEAT + lane * 4);
        float4 r1 = *(const float4*)(h + (size_t)s1 * NFEAT + lane * 4);
        float4 r2 = *(const float4*)(h + (size_t)s2 * NFEAT + lane * 4);
        float4 r3 = *(const float4*)(h + (size_t)s3 * NFEAT + lane * 4);
        acc.x += r0.x; acc.y += r0.y; acc.z += r0.z; acc.w += r0.w;
        acc.x += r1.x; acc.y += r1.y; acc.z += r1.z; acc.w += r1.w;
        acc.x += r2.x; acc.y += r2.y; acc.z += r2.z; acc.w += r2.w;
        acc.x += r3.x; acc.y += r3.y; acc.z += r3.z; acc.w += r3.w;
    }

    float nrm = norm[n];
    float g   = gate[n];
    float4 own = *(const float4*)(h + (size_t)n * NFEAT + lane * 4);
    float4 o;
    o.x = own.x + g * fmaxf(acc.x * nrm, 0.0f);
    o.y = own.y + g * fmaxf(acc.y * nrm, 0.0f);
    o.z = own.z + g * fmaxf(acc.z * nrm, 0.0f);
    o.w = own.w + g * fmaxf(acc.w * nrm, 0.0f);
    *(float4*)(new_h + (size_t)n * NFEAT + lane * 4) = o;
}

// ---------------------------------------------------------------- launcher
extern "C" void kernel_launch(void* const* d_in, const int* in_sizes, int n_in,
                              void* d_out, int out_size, void* d_ws, size_t ws_size,
                              hipStream_t stream) {
    const float* h      = (const float*)d_in[0];
    const float* logits = (const float*)d_in[1];
    const float* old_z  = (const float*)d_in[2];
    const float* norm   = (const float*)d_in[3];
    // d_in[4] = in_degs (constant 32, unused)
    const float* tau1   = (const float*)d_in[5];
    const float* tau2   = (const float*)d_in[6];
    const int*   src    = (const int*)d_in[7];
    // d_in[8] = dst (uniform: dst[e] = e / DEG, unused)

    const int n = in_sizes[2];               // N_NODES (old_z length)

    float* out_h = (float*)d_out;                     // [N, 128]
    float* out_z = out_h + (size_t)n * NFEAT;         // [N]

    char*  ws       = (char*)d_ws;
    int*   pred     = (int*)ws;                                 // N ints
    float* f1       = (float*)(ws + (size_t)n * 4);             // N floats
    float* f2       = (float*)(ws + (size_t)n * 8);             // N floats
    float* gate     = (float*)(ws + (size_t)n * 12);            // N floats
    float* partials = (float*)(ws + (size_t)n * 16);            // 4*NB_RED
    float* stats    = partials + 4 * NB_RED;                    // 4 floats

    int tpb = 256;
    int wavesPerBlock = tpb / 32;

    pred_kernel   <<<(n + tpb - 1) / tpb, tpb, 0, stream>>>(logits, pred, n);
    f1f2_kernel   <<<(n + wavesPerBlock - 1) / wavesPerBlock, tpb, 0, stream>>>(
                      pred, src, f1, f2, n);
    partial_kernel<<<NB_RED, tpb, 0, stream>>>(f1, f2, partials, n);
    stats_kernel  <<<1, 32, 0, stream>>>(partials, stats, n);
    zgate_kernel  <<<(n + tpb - 1) / tpb, tpb, 0, stream>>>(
                      f1, f2, old_z, tau1, tau2, stats, out_z, gate, n);
    agg_kernel    <<<(n + wavesPerBlock - 1) / wavesPerBlock, tpb, 0, stream>>>(
                      h, src, norm, gate, out_h, n);
}